// GAGRUModel_47691316855469
// MI455X (gfx1250) — compile-verified
//
#include <hip/hip_runtime.h>
#include <math.h>

// Problem constants from the reference
#define B_ 16
#define T_ 8
#define F_ 2
#define N_ 1024
#define H_ 64
#define HOR_ 4
#define SLOPE_ 0.2f
#define LDP 66   // padded LDS row stride (floats): 264B -> 8B aligned, conflict-free

typedef __attribute__((ext_vector_type(2))) float v2f;
typedef __attribute__((ext_vector_type(8))) float v8f;

struct GParams {
  const float* W_in[3];
  const float* b_in[3];
  const float* a_self_in[3];
  const float* a_nb_in[3];
  const float* W_h[3];
  const float* b_h[3];
  const float* a_self_h[3];
  const float* a_nb_h[3];
  const float* fc_W;
  const float* fc_b;
};

__device__ __forceinline__ float lrelu(float v) { return v > 0.f ? v : SLOPE_ * v; }

// ------------------------------------------------------------------
// Fold W*a_self / W*a_nb and b.a scalars once per launch.
// wa[idx][c], bsc[idx]; idx = (g*2+src)*2+which, src: 0=in(C=2),1=h(C=64)
// ------------------------------------------------------------------
__global__ void prep_kernel(GParams P, float* wa, float* bsc) {
  int c = threadIdx.x;  // 0..63
  for (int g = 0; g < 3; ++g) {
    {
      const float* W = P.W_h[g];
      const float* as = P.a_self_h[g];
      const float* an = P.a_nb_h[g];
      float s0 = 0.f, s1 = 0.f;
      for (int h = 0; h < H_; ++h) { float w = W[c * H_ + h]; s0 = fmaf(w, as[h], s0); s1 = fmaf(w, an[h], s1); }
      wa[((g * 2 + 1) * 2 + 0) * 64 + c] = s0;
      wa[((g * 2 + 1) * 2 + 1) * 64 + c] = s1;
    }
    if (c < F_) {
      const float* W = P.W_in[g];
      float s0 = 0.f, s1 = 0.f;
      for (int h = 0; h < H_; ++h) { float w = W[c * H_ + h]; s0 = fmaf(w, P.a_self_in[g][h], s0); s1 = fmaf(w, P.a_nb_in[g][h], s1); }
      wa[((g * 2 + 0) * 2 + 0) * 64 + c] = s0;
      wa[((g * 2 + 0) * 2 + 1) * 64 + c] = s1;
    }
    if (c == 0) {
      float d_si = 0, d_ni = 0, d_sh = 0, d_nh = 0;
      for (int h = 0; h < H_; ++h) {
        d_si = fmaf(P.b_in[g][h], P.a_self_in[g][h], d_si);
        d_ni = fmaf(P.b_in[g][h], P.a_nb_in[g][h], d_ni);
        d_sh = fmaf(P.b_h[g][h], P.a_self_h[g][h], d_sh);
        d_nh = fmaf(P.b_h[g][h], P.a_nb_h[g][h], d_nh);
      }
      bsc[(g * 2 + 0) * 2 + 0] = d_si;
      bsc[(g * 2 + 0) * 2 + 1] = d_ni;
      bsc[(g * 2 + 1) * 2 + 0] = d_sh;
      bsc[(g * 2 + 1) * 2 + 1] = d_nh;
    }
  }
}

// ------------------------------------------------------------------
// s_self / s_nb for all 12 (gate,src,which) combos: s = feat^T (W a) + b.a
// ------------------------------------------------------------------
__global__ void scores_kernel(const float* featbase, int feat_bstride,
                              const float* hid, const float* wa, const float* bsc,
                              float* sbuf) {
  int idx = blockIdx.z;            // (g*2+src)*2+which
  int src = (idx >> 1) & 1;
  int b = blockIdx.y;
  int n = blockIdx.x * blockDim.x + threadIdx.x;
  int C = src ? H_ : F_;
  const float* f = src ? (hid + (size_t)b * H_ * N_)
                       : (featbase + (size_t)b * feat_bstride);
  const float* w = wa + idx * 64;
  float acc = bsc[idx];
  for (int c = 0; c < C; ++c) acc = fmaf(f[c * N_ + n], w[c], acc);
  sbuf[(size_t)idx * B_ * N_ + b * N_ + n] = acc;
}

// ------------------------------------------------------------------
// GAT for the C=2 input features: P (2 rows) + colsum + tiny K=2 output GEMM.
// VALU-bound (dominated by the per-(n,m) exp), one thread per output column m.
// ------------------------------------------------------------------
__global__ void gat_in_kernel(const float* featbase, int feat_bstride,
                              const float* mask, const float* sbuf,
                              GParams P, float* Yi) {
  __shared__ float f0[N_], f1[N_], ss[N_];
  __shared__ float w0[H_], w1[H_], bi[H_];
  int g = blockIdx.z, b = blockIdx.y;
  int tid = threadIdx.x;
  int m = blockIdx.x * 128 + tid;
  const float* fb = featbase + (size_t)b * feat_bstride;
  const float* ssp = sbuf + (size_t)((g * 2 + 0) * 2 + 0) * B_ * N_ + b * N_;
  const float* snp = sbuf + (size_t)((g * 2 + 0) * 2 + 1) * B_ * N_ + b * N_;
  for (int i = tid; i < N_; i += 128) { f0[i] = fb[i]; f1[i] = fb[N_ + i]; ss[i] = ssp[i]; }
  for (int i = tid; i < H_; i += 128) { w0[i] = P.W_in[g][i]; w1[i] = P.W_in[g][H_ + i]; bi[i] = P.b_in[g][i]; }
  __syncthreads();
  float sb = snp[m];
  float a0 = 0.f, a1 = 0.f, cs = 0.f;
  for (int n = 0; n < N_; ++n) {
    float sc = lrelu(ss[n] + sb);
    // E = 1 where mask==0, exp(score) where mask==1 (scores masked, not probs)
    float E = fmaf(mask[(size_t)n * N_ + m], __expf(sc) - 1.f, 1.f);
    a0 = fmaf(f0[n], E, a0);
    a1 = fmaf(f1[n], E, a1);
    cs += E;
  }
  float inv = 1.f / cs;
  a0 *= inv; a1 *= inv;
  float* y = Yi + (((size_t)g * B_ + b) * N_ + m) * H_;
  for (int h = 0; h < H_; ++h) y[h] = fmaf(a0, w0[h], fmaf(a1, w1[h], bi[h]));
}

// ------------------------------------------------------------------
// GAT for the C=64 hidden features — the heavy path, fully WMMA (f32 16x16x4).
// Block: 128 threads (4 waves), owns all 64 rows x 32 output columns.
// Stage1: P = feat x E (K=1024, E tiles generated in LDS on the fly,
//         colsum accumulated from the same tiles).
// Stage2: Y = (P/colsum)^T x W + b  (K=64), same block, no global round-trip.
// WMMA f32 16x16x4 VGPR maps used:
//   A/B frag: lane L, vgpr v -> row = L&15, K = v + 2*(L>>4)  (float2 ds_load_b64)
//   C/D:      lane L, vgpr v -> M = v + 8*(L>>4), N = L&15
// ------------------------------------------------------------------
__global__ void __launch_bounds__(128)
gat_h_kernel(const float* hid, const float* mask, const float* sbuf,
             GParams P, float* Yh) {
  __shared__ float Ald[64 * LDP];  // stage1 A (c x k); reused as W^T (h x c) in stage2
  __shared__ float Eld[32 * LDP];  // stage1 B (m x k); reused as Agg  (m x c) in stage2
  __shared__ float csld[32];
  int g = blockIdx.z, b = blockIdx.y;
  int m0 = blockIdx.x * 32;
  int tid = threadIdx.x;
  int lane = tid & 31, wv = tid >> 5;
  int nlo = lane & 15;
  int hi2 = (lane >> 4) << 1;

  const float* feat = hid + (size_t)b * H_ * N_;
  const float* ssp = sbuf + (size_t)((g * 2 + 1) * 2 + 0) * B_ * N_ + b * N_;
  const float* snp = sbuf + (size_t)((g * 2 + 1) * 2 + 1) * B_ * N_ + b * N_;

  if (tid < 32) csld[tid] = 0.f;
  float snb_v = snp[m0 + (tid & 31)];

  v8f acc0 = {};  // wave wv owns c-rows [16*wv, 16*wv+16); acc0 -> m cols [m0,m0+16)
  v8f acc1 = {};  // acc1 -> m cols [m0+16, m0+32)
  const float* Arow = &Ald[(wv * 16 + nlo) * LDP];
  const float* Brow0 = &Eld[nlo * LDP];
  const float* Brow1 = &Eld[(16 + nlo) * LDP];

  for (int k0 = 0; k0 < N_; k0 += 64) {
    __syncthreads();  // previous iter's WMMA reads done before refill
    // A tile: feat[c][k0..k0+64)
    for (int i = tid; i < 64 * 64; i += 128) {
      int c = i >> 6, kl = i & 63;
      Ald[c * LDP + kl] = feat[c * N_ + k0 + kl];
    }
    // E tile (transposed: [m_local][k]) + colsum partials
    {
      int ml = tid & 31;
      int mg = m0 + ml;
      float part = 0.f;
      for (int kl = tid >> 5; kl < 64; kl += 4) {
        float sc = lrelu(ssp[k0 + kl] + snb_v);
        float E = fmaf(mask[(size_t)(k0 + kl) * N_ + mg], __expf(sc) - 1.f, 1.f);
        Eld[ml * LDP + kl] = E;
        part += E;
      }
      atomicAdd(&csld[ml], part);  // ds_add_f32
    }
    if (k0 + 64 < N_)  // pull next mask K-tile toward L2/L0 while WMMA runs
      __builtin_prefetch(&mask[(size_t)(k0 + 64 + (tid >> 5)) * N_ + m0 + (tid & 31)], 0, 1);
    __syncthreads();
#pragma unroll
    for (int ks = 0; ks < 16; ++ks) {
      int kb = ks * 4 + hi2;
      v2f a  = *(const v2f*)(Arow + kb);
      v2f b0 = *(const v2f*)(Brow0 + kb);
      v2f b1 = *(const v2f*)(Brow1 + kb);
      acc0 = __builtin_amdgcn_wmma_f32_16x16x4_f32(false, a, false, b0, (short)0, acc0, false, false);
      acc1 = __builtin_amdgcn_wmma_f32_16x16x4_f32(false, a, false, b1, (short)0, acc1, false, false);
    }
  }
  __syncthreads();

  // Agg = P / colsum, stored transposed [m_local][c] into Eld (reuse)
#pragma unroll
  for (int v = 0; v < 8; ++v) {
    int c = 16 * wv + v + 8 * (lane >> 4);
    Eld[nlo * LDP + c]        = acc0[v] / csld[nlo];
    Eld[(16 + nlo) * LDP + c] = acc1[v] / csld[16 + nlo];
  }
  // W^T into Ald (reuse): Ald[h][c] = W[c][h]
  {
    const float* W = P.W_h[g];
    for (int i = tid; i < 64 * 64; i += 128) {
      int c = i >> 6, h = i & 63;
      Ald[h * LDP + c] = W[c * H_ + h];
    }
  }
  __syncthreads();

  // Stage2: Y[m][h] = Agg[m][:] x W[:][h]; wave wv owns h-tile wv, both m-tiles
  v8f y0 = {};
  v8f y1 = {};
  const float* A0 = &Eld[nlo * LDP];
  const float* A1 = &Eld[(16 + nlo) * LDP];
  const float* Bw = &Ald[(16 * wv + nlo) * LDP];
#pragma unroll
  for (int ks = 0; ks < 16; ++ks) {
    int kb = ks * 4 + hi2;
    v2f bb = *(const v2f*)(Bw + kb);
    v2f a0 = *(const v2f*)(A0 + kb);
    v2f a1 = *(const v2f*)(A1 + kb);
    y0 = __builtin_amdgcn_wmma_f32_16x16x4_f32(false, a0, false, bb, (short)0, y0, false, false);
    y1 = __builtin_amdgcn_wmma_f32_16x16x4_f32(false, a1, false, bb, (short)0, y1, false, false);
  }
  float bv = P.b_h[g][16 * wv + nlo];
  float* Yp = Yh + ((size_t)g * B_ + b) * N_ * H_;
#pragma unroll
  for (int v = 0; v < 8; ++v) {
    int mA = m0 + v + 8 * (lane >> 4);
    int mB = mA + 16;
    Yp[(size_t)mA * H_ + 16 * wv + nlo] = y0[v] + bv;
    Yp[(size_t)mB * H_ + 16 * wv + nlo] = y1[v] + bv;
  }
}

// ------------------------------------------------------------------
// GRU combine: r,u = sigmoid(gh+gi); n = tanh(r*gh_new + gi_new);
// h' = (1-u)*n + u*h  (output back in [B,H,N] layout)
// ------------------------------------------------------------------
__global__ void combine_kernel(const float* Yh, const float* Yi,
                               const float* hold, float* hnew) {
  int b = blockIdx.y;
  int i = blockIdx.x * blockDim.x + threadIdx.x;  // over N*H
  int h = i & (H_ - 1);
  int m = i >> 6;
  size_t base = ((size_t)b * N_ + m) * H_ + h;
  size_t gstr = (size_t)B_ * N_ * H_;
  float rs = Yh[0 * gstr + base] + Yi[0 * gstr + base];
  float us = Yh[1 * gstr + base] + Yi[1 * gstr + base];
  float r = 1.f / (1.f + __expf(-rs));
  float u = 1.f / (1.f + __expf(-us));
  float nn = tanhf(fmaf(r, Yh[2 * gstr + base], Yi[2 * gstr + base]));
  size_t hix = ((size_t)b * H_ + h) * N_ + m;
  hnew[hix] = fmaf(u, hold[hix], (1.f - u) * nn);
}

// ------------------------------------------------------------------
// Decoder head: out[b,m] = sum_h hidden[b,h,m]*fcW[h] + fcb; feeds back as feat
// ------------------------------------------------------------------
__global__ void dec_out_kernel(const float* hid, const float* fcW, const float* fcb,
                               float* out, float* featbuf, int step) {
  __shared__ float fw[H_];
  int b = blockIdx.y;
  int tid = threadIdx.x;
  int m = blockIdx.x * 256 + tid;
  if (tid < H_) fw[tid] = fcW[tid];
  __syncthreads();
  const float* hb = hid + (size_t)b * H_ * N_;
  float acc = fcb[0];
  for (int h = 0; h < H_; ++h) acc = fmaf(hb[h * N_ + m], fw[h], acc);
  out[((size_t)b * HOR_ + step) * N_ + m] = acc;
  featbuf[(size_t)b * F_ * N_ + m] = acc;
  featbuf[(size_t)b * F_ * N_ + N_ + m] = 0.f;
}

// ------------------------------------------------------------------
extern "C" void kernel_launch(void* const* d_in, const int* in_sizes, int n_in,
                              void* d_out, int out_size, void* d_ws, size_t ws_size,
                              hipStream_t stream) {
  (void)in_sizes; (void)n_in; (void)out_size; (void)ws_size;
  const float* x = (const float*)d_in[0];
  const float* hidden0 = (const float*)d_in[1];
  const float* mask = (const float*)d_in[2];
  GParams P;
  for (int g = 0; g < 3; ++g) {
    const int base = 3 + 8 * g;
    P.W_in[g]      = (const float*)d_in[base + 0];
    P.b_in[g]      = (const float*)d_in[base + 1];
    P.a_self_in[g] = (const float*)d_in[base + 2];
    P.a_nb_in[g]   = (const float*)d_in[base + 3];
    P.W_h[g]       = (const float*)d_in[base + 4];
    P.b_h[g]       = (const float*)d_in[base + 5];
    P.a_self_h[g]  = (const float*)d_in[base + 6];
    P.a_nb_h[g]    = (const float*)d_in[base + 7];
  }
  P.fc_W = (const float*)d_in[27];
  P.fc_b = (const float*)d_in[28];
  float* out = (float*)d_out;

  // Workspace partition (~34.5 MB of f32)
  float* wsf = (float*)d_ws;
  float* wa = wsf;                              // 768
  float* bsc = wa + 768;                        // 16
  float* sbuf = bsc + 16;                       // 12*B*N
  float* Yh = sbuf + (size_t)12 * B_ * N_;      // 3*B*N*H
  float* Yi = Yh + (size_t)3 * B_ * N_ * H_;    // 3*B*N*H
  float* hidA = Yi + (size_t)3 * B_ * N_ * H_;  // B*H*N
  float* hidB = hidA + (size_t)B_ * H_ * N_;    // B*H*N
  float* featbuf = hidB + (size_t)B_ * H_ * N_; // B*F*N

  hipMemcpyAsync(hidA, hidden0, (size_t)B_ * H_ * N_ * sizeof(float),
                 hipMemcpyDeviceToDevice, stream);
  prep_kernel<<<dim3(1), dim3(64), 0, stream>>>(P, wa, bsc);

  float* cur = hidA;
  float* nxt = hidB;
  for (int t = 0; t < T_ + HOR_; ++t) {
    const float* featbase;
    int bstride;
    if (t < T_) {                       // encoder: feat = x[:, t]
      featbase = x + (size_t)t * F_ * N_;
      bstride = T_ * F_ * N_;
    } else {                            // decoder: out from current h feeds back
      dec_out_kernel<<<dim3(N_ / 256, B_), 256, 0, stream>>>(
          cur, P.fc_W, P.fc_b, out, featbuf, t - T_);
      featbase = featbuf;
      bstride = F_ * N_;
    }
    scores_kernel<<<dim3(N_ / 256, B_, 12), 256, 0, stream>>>(
        featbase, bstride, cur, wa, bsc, sbuf);
    gat_in_kernel<<<dim3(N_ / 128, B_, 3), 128, 0, stream>>>(
        featbase, bstride, mask, sbuf, P, Yi);
    gat_h_kernel<<<dim3(N_ / 32, B_, 3), 128, 0, stream>>>(
        cur, mask, sbuf, P, Yh);
    combine_kernel<<<dim3(N_ * H_ / 256, B_), 256, 0, stream>>>(Yh, Yi, cur, nxt);
    float* tmp = cur; cur = nxt; nxt = tmp;
  }
}